// AttentionWithBiasAndExtraOutput_51866025066592
// MI455X (gfx1250) — compile-verified
//
#include <hip/hip_runtime.h>

#define DIM     1024
#define NHEADS  16
#define HDIM    64
#define BATCH   4
#define SEQ     1024
#define SCALEF  0.125f   // 64^-0.5

typedef float v2f __attribute__((ext_vector_type(2)));
typedef float v8f __attribute__((ext_vector_type(8)));
typedef unsigned int u32x4 __attribute__((ext_vector_type(4)));
typedef int i32x4 __attribute__((ext_vector_type(4)));
typedef int i32x8 __attribute__((ext_vector_type(8)));

// A/B fragment for V_WMMA_F32_16X16X4_F32.
// lanes 0-15: row = row0+lane, K = k0+{0,1}; lanes 16-31: row = row0+lane-16, K = k0+{2,3}
__device__ __forceinline__ v2f ld_frag(const float* __restrict__ base, int row0, int k0, int ld) {
    const int lane = threadIdx.x & 31;
    const float* p = base + (size_t)(row0 + (lane & 15)) * ld + (k0 + ((lane >> 4) << 1));
    v2f f; f.x = p[0]; f.y = p[1];
    return f;
}

__device__ __forceinline__ v8f wmma_f32(v2f a, v2f b, v8f c) {
    return __builtin_amdgcn_wmma_f32_16x16x4_f32(false, a, false, b, (short)0, c, false, false);
}

// LDS byte address of a generic pointer into __shared__ (ISA: flat->LDS uses addr[31:0])
__device__ __forceinline__ unsigned lds_off(const void* p) {
    return (unsigned)(unsigned long long)p;
}

// Stage 32KB (128 rows x 64 f32, contiguous) global -> LDS via CDNA5 async
// vector copies: 256 threads x 8 iters x 16B = 32KB, fully coalesced.
__device__ __forceinline__ void stage32k_async(const float* __restrict__ srcBase, unsigned ldsBase) {
    const float* src = srcBase + threadIdx.x * 4;   // 16B per lane
    unsigned dst = ldsBase + threadIdx.x * 16;
#pragma unroll
    for (int i = 0; i < 8; ++i) {
        asm volatile("global_load_async_to_lds_b128 %0, %1, off"
                     :: "v"(dst + i * 4096), "v"(src + (size_t)i * 1024)
                     : "memory");
    }
    asm volatile("s_wait_asynccnt 0x0" ::: "memory");
}

// Stage a rows x 64 f32 tile global -> LDS with ONE Tensor Data Mover descriptor
// (D# per CDNA5 ISA). Issue from a single wave.
__device__ __forceinline__ void stage_tdm_2d(const void* gsrc, unsigned ldsByteAddr, unsigned rows) {
    const unsigned long long ga = (unsigned long long)gsrc;
    u32x4 g0;
    g0.x = 1u;                                                  // count=1, user descriptor
    g0.y = ldsByteAddr;                                         // lds_addr  [63:32]
    g0.z = (unsigned)ga;                                        // global_addr[95:64]
    g0.w = (unsigned)((ga >> 32) & 0x01FFFFFFu) | (2u << 30);   // addr[56:32] + type=2
    i32x8 g1;
    g1[0] = (int)(2u << 16);          // workgroup_mask=0, data_size=2 (4 bytes)
    g1[1] = (int)(64u << 16);         // tensor_dim0 = 64 (bits 79:48, low16)
    g1[2] = (int)(rows << 16);        // tensor_dim1 = rows (bits 111:80, low16)
    g1[3] = (int)(64u << 16);         // tile_dim0 = 64 (bits 127:112)
    g1[4] = (int)rows;                // tile_dim1 = rows (bits 143:128)
    g1[5] = 64;                       // tensor_dim0_stride = 64 elems (bits 207:160)
    g1[6] = 0;
    g1[7] = 0;
    const i32x4 z4 = {0, 0, 0, 0};    // groups 2/3 unused (2-D tensor)
    const i32x8 z8 = {0, 0, 0, 0, 0, 0, 0, 0};
    __builtin_amdgcn_tensor_load_to_lds(g0, g1, z4, z4, z8, 0);
}

// ---------------------------------------------------------------------------
// Kernel 1: head-layout projection  dst(B,H,L,D) = X(B*L, DIM) @ W(DIM,DIM)^T
// ---------------------------------------------------------------------------
__global__ void __launch_bounds__(256) proj_gemm(const float* __restrict__ X,
                                                 const float* __restrict__ W,
                                                 float* __restrict__ dstHead) {
    const int wave = threadIdx.x >> 5;
    const int lane = threadIdx.x & 31;
    const int m0 = blockIdx.x * 128 + wave * 16;
    const int n0 = blockIdx.y * 64;

    v8f acc[4] = {};
    for (int k0 = 0; k0 < DIM; k0 += 4) {
        if ((k0 & 63) == 0 && k0 + 64 < DIM)   // global_prefetch_b8 of next A chunk
            __builtin_prefetch(X + (size_t)(m0 + (lane & 15)) * DIM + k0 + 64, 0, 1);
        v2f a = ld_frag(X, m0, k0, DIM);
#pragma unroll
        for (int t = 0; t < 4; ++t) {
            v2f b = ld_frag(W, n0 + 16 * t, k0, DIM);
            acc[t] = wmma_f32(a, b, acc[t]);
        }
    }
    const int rb   = (lane >> 4) << 3;   // 0 or 8
    const int ncol = lane & 15;
#pragma unroll
    for (int t = 0; t < 4; ++t) {
        const int n = n0 + 16 * t + ncol;
        const int h = n >> 6, d = n & 63;
#pragma unroll
        for (int v = 0; v < 8; ++v) {
            const int m = m0 + rb + v;
            const int b = m >> 10, l = m & 1023;
            dstHead[(((size_t)(b * NHEADS + h) << 10) + l) * HDIM + d] = acc[t][v];
        }
    }
}

// accumulate one 16x16 tile of attn@V over a 128-deep K chunk (A from LDS scores,
// B from LDS-staged V chunk)
__device__ __forceinline__ v8f attn_tile_acc(v8f acc, const float* __restrict__ sc,
                                             const float* __restrict__ vstage,
                                             int tr, int tc, int k0c, int lane) {
    const int koff = (lane >> 4) << 1;
    const int arow = tr * 16 + (lane & 15);
    const int bcol = tc * 16 + (lane & 15);
#pragma unroll 4
    for (int kl = 0; kl < 128; kl += 4) {
        const float* ap = sc + arow * SEQ + (k0c + kl + koff);
        v2f a; a.x = ap[0]; a.y = ap[1];
        const float* bp = vstage + (kl + koff) * HDIM + bcol;
        v2f bf; bf.x = bp[0]; bf.y = bp[HDIM];
        acc = wmma_f32(a, bf, acc);
    }
    return acc;
}

// ---------------------------------------------------------------------------
// Kernel 2: fused attention for one (b,h) and a 64-row query block.
//   scores = Q K^T * scale     -> qk output (pre-bias), tile kept in LDS
//   p      = softmax(scores+b) -> in-place in LDS (64x1024 f32 = 256KB)
//   X      = p @ V             -> context to workspace (B,N,DIM)
// K staged via async-to-LDS vector copies; V staged via TDM descriptors.
// ---------------------------------------------------------------------------
__global__ void __launch_bounds__(256) attn_kernel(const float* __restrict__ Qh,
                                                   const float* __restrict__ Kh,
                                                   const float* __restrict__ Vh,
                                                   const float* __restrict__ bias,
                                                   float* __restrict__ qk_out,
                                                   float* __restrict__ Xctx) {
    extern __shared__ float smem[];
    float* sc      = smem;                    // 64 x 1024 score/prob tile
    float* red     = smem + 64 * SEQ;         // 64 x 4 reduction scratch
    float* kvstage = red + 256;               // 128 x 64 K/V staging (32KB)
    const unsigned stageBase = lds_off(kvstage);

    const int bh = blockIdx.x;                // 0..63
    const int b  = bh >> 4, h = bh & 15;
    const int q0 = blockIdx.y * 64;           // query row block start
    const int wave = threadIdx.x >> 5;
    const int lane = threadIdx.x & 31;

    const float* Qb  = Qh   + (size_t)bh * SEQ * HDIM;
    const float* Kb  = Kh   + (size_t)bh * SEQ * HDIM;
    const float* Vb  = Vh   + (size_t)bh * SEQ * HDIM;
    const float* Bb  = bias + (size_t)bh * SEQ * SEQ;
    float*       qkb = qk_out + (size_t)bh * SEQ * SEQ;

    // ---- Pass 1: scores = Q K^T * scale -------------------------------
    const int r     = wave & 3;               // row-tile 0..3 (16 rows each)
    const int chalf = wave >> 2;              // column half within 128-col chunk

    v2f afrag[16];                            // whole K=64 Q strip in registers
#pragma unroll
    for (int kk = 0; kk < 16; ++kk)
        afrag[kk] = ld_frag(Qb, q0 + r * 16, kk * 4, HDIM);

    for (int c0 = 0; c0 < SEQ; c0 += 128) {
        stage32k_async(Kb + (size_t)c0 * HDIM, stageBase);   // 128 key rows -> LDS
        __syncthreads();
#pragma unroll
        for (int ct = 0; ct < 4; ++ct) {
            const int nloc = chalf * 64 + ct * 16;           // within chunk
            v8f acc = {};
#pragma unroll
            for (int kk = 0; kk < 16; ++kk) {
                v2f bf = ld_frag(kvstage, nloc, kk * 4, HDIM);   // LDS read
                acc = wmma_f32(afrag[kk], bf, acc);
            }
            const int ncol = c0 + nloc + (lane & 15);
            const int rbo  = (lane >> 4) << 3;
#pragma unroll
            for (int v = 0; v < 8; ++v) {
                const int ml = r * 16 + rbo + v;             // local row 0..63
                const float s = acc[v] * SCALEF;
                qkb[(size_t)(q0 + ml) * SEQ + ncol] = s;
                sc[ml * SEQ + ncol] = s + Bb[(size_t)(q0 + ml) * SEQ + ncol];
            }
        }
        __syncthreads();
    }

    // ---- softmax over each of the 64 rows (4 threads / row) -----------
    {
        const int row  = threadIdx.x >> 2;
        const int part = threadIdx.x & 3;
        float* rp = sc + row * SEQ + part * 256;

        float m = -3.0e38f;
        for (int i = 0; i < 256; ++i) m = fmaxf(m, rp[i]);
        red[row * 4 + part] = m;
        __syncthreads();
        const float rm = fmaxf(fmaxf(red[row * 4 + 0], red[row * 4 + 1]),
                               fmaxf(red[row * 4 + 2], red[row * 4 + 3]));

        float sum = 0.f;
        for (int i = 0; i < 256; ++i) { float e = __expf(rp[i] - rm); rp[i] = e; sum += e; }
        __syncthreads();                      // all reads of max done before reuse
        red[row * 4 + part] = sum;
        __syncthreads();
        const float rs = red[row * 4 + 0] + red[row * 4 + 1] +
                         red[row * 4 + 2] + red[row * 4 + 3];
        const float inv = __builtin_amdgcn_rcpf(rs);
        for (int i = 0; i < 256; ++i) rp[i] *= inv;
    }
    __syncthreads();

    // ---- Pass 2: X = attn @ V, V staged via Tensor Data Mover ----------
    const int tr0 = (wave * 2) >> 2,     tc0 = (wave * 2) & 3;
    const int tr1 = (wave * 2 + 1) >> 2, tc1 = (wave * 2 + 1) & 3;
    v8f acc0 = {}, acc1 = {};
    for (int k0c = 0; k0c < SEQ; k0c += 128) {
        if (threadIdx.x < 32) {               // one wave issues the tile DMA
            stage_tdm_2d(Vb + (size_t)k0c * HDIM, stageBase, 128);
            __builtin_amdgcn_s_wait_tensorcnt(0);
        }
        __syncthreads();                      // tile visible to all waves
        acc0 = attn_tile_acc(acc0, sc, kvstage, tr0, tc0, k0c, lane);
        acc1 = attn_tile_acc(acc1, sc, kvstage, tr1, tc1, k0c, lane);
        __syncthreads();                      // before next chunk overwrites stage
    }
    {
        const int rbo = (lane >> 4) << 3;
#pragma unroll
        for (int v = 0; v < 8; ++v) {
            const int l0 = q0 + tr0 * 16 + rbo + v;
            Xctx[((size_t)(b * SEQ + l0)) * DIM + h * HDIM + tc0 * 16 + (lane & 15)] = acc0[v];
            const int l1 = q0 + tr1 * 16 + rbo + v;
            Xctx[((size_t)(b * SEQ + l1)) * DIM + h * HDIM + tc1 * 16 + (lane & 15)] = acc1[v];
        }
    }
}

// ---------------------------------------------------------------------------
// Kernel 3: out(B*N, DIM) = X @ Wo^T + bo
// ---------------------------------------------------------------------------
__global__ void __launch_bounds__(256) out_gemm(const float* __restrict__ X,
                                                const float* __restrict__ Wo,
                                                const float* __restrict__ bo,
                                                float* __restrict__ out) {
    const int wave = threadIdx.x >> 5;
    const int lane = threadIdx.x & 31;
    const int m0 = blockIdx.x * 128 + wave * 16;
    const int n0 = blockIdx.y * 64;

    v8f acc[4] = {};
    for (int k0 = 0; k0 < DIM; k0 += 4) {
        if ((k0 & 63) == 0 && k0 + 64 < DIM)
            __builtin_prefetch(X + (size_t)(m0 + (lane & 15)) * DIM + k0 + 64, 0, 1);
        v2f a = ld_frag(X, m0, k0, DIM);
#pragma unroll
        for (int t = 0; t < 4; ++t) {
            v2f b = ld_frag(Wo, n0 + 16 * t, k0, DIM);
            acc[t] = wmma_f32(a, b, acc[t]);
        }
    }
    const int rb = (lane >> 4) << 3;
#pragma unroll
    for (int t = 0; t < 4; ++t) {
        const int n = n0 + 16 * t + (lane & 15);
        const float bb = bo[n];
#pragma unroll
        for (int v = 0; v < 8; ++v) {
            const int m = m0 + rb + v;
            out[(size_t)m * DIM + n] = acc[t][v] + bb;
        }
    }
}

extern "C" void kernel_launch(void* const* d_in, const int* in_sizes, int n_in,
                              void* d_out, int out_size, void* d_ws, size_t ws_size,
                              hipStream_t stream) {
    (void)in_sizes; (void)n_in; (void)out_size; (void)ws_size;

    const float* q    = (const float*)d_in[0];
    const float* k    = (const float*)d_in[1];
    const float* v    = (const float*)d_in[2];
    const float* bias = (const float*)d_in[3];
    const float* Wq   = (const float*)d_in[4];
    const float* Wk   = (const float*)d_in[5];
    const float* Wv   = (const float*)d_in[6];
    const float* Wo   = (const float*)d_in[7];
    const float* bo   = (const float*)d_in[8];

    float* out = (float*)d_out;                            // (B,N,DIM)
    float* qk  = out + (size_t)BATCH * SEQ * DIM;          // (B,H,N,S)

    const size_t headElems = (size_t)BATCH * NHEADS * SEQ * HDIM;  // 4M floats
    float* Qh = (float*)d_ws;
    float* Kh = Qh + headElems;
    float* Vh = Kh + headElems;
    float* Xc = Vh + headElems;                            // (B,N,DIM) context

    dim3 blk(256);
    dim3 g1(BATCH * SEQ / 128, DIM / 64);                  // 32 x 16

    proj_gemm<<<g1, blk, 0, stream>>>(q, Wq, Qh);
    proj_gemm<<<g1, blk, 0, stream>>>(k, Wk, Kh);
    proj_gemm<<<g1, blk, 0, stream>>>(v, Wv, Vh);

    dim3 g2(BATCH * NHEADS, SEQ / 64);                     // 64 x 16
    size_t smem = (size_t)(64 * SEQ + 256 + 128 * HDIM) * sizeof(float);  // ~289 KB
    attn_kernel<<<g2, blk, smem, stream>>>(Qh, Kh, Vh, bias, qk, Xc);

    out_gemm<<<g1, blk, 0, stream>>>(Xc, Wo, bo, out);
}